// FeatureAttentionLayer_41240275976732
// MI455X (gfx1250) — compile-verified
//
#include <hip/hip_runtime.h>
#include <math.h>

#define B_ 32
#define W_ 128
#define K_ 128
#define E_ 256
#define NEG_SLOPE 0.2f

typedef float v2f __attribute__((ext_vector_type(2)));
typedef float v8f __attribute__((ext_vector_type(8)));

// ---------------------------------------------------------------------------
// Kernel A: L[b,k,e] = sum_w x[b,w,k] * lin_w[e, w]
//           R[b,k,e] = sum_w x[b,w,k] * lin_w[e, W + w]
// One wave per 16x16 (k,e) output tile, V_WMMA_F32_16X16X4_F32, inner dim W.
// Tiles: b(32) * ktile(8) * ntile(32: 16 for L, 16 for R) = 8192 waves.
// ---------------------------------------------------------------------------
__global__ __launch_bounds__(128) void gat_proj_kernel(
    const float* __restrict__ x, const float* __restrict__ lin_w,
    float* __restrict__ Lp, float* __restrict__ Rp) {
  const int wave = threadIdx.x >> 5;
  const int lane = threadIdx.x & 31;
  const int tile = blockIdx.x * 4 + wave;
  const int b   = tile >> 8;          // / (8*32)
  const int kt  = (tile >> 5) & 7;
  const int nt  = tile & 31;
  const int isR = nt >> 4;
  const int e0  = (nt & 15) << 4;
  const int k0  = kt << 4;
  const int wOff = isR ? W_ : 0;

  const int m  = lane & 15;           // A: M row / B: N col
  const int kk = (lane >> 4) << 1;    // K sub-offset {0,2}

  const float* xb   = x + b * (W_ * K_);
  const float* wrow = lin_w + (e0 + m) * (2 * W_) + wOff;

  v8f c = {};
#pragma unroll 8
  for (int s = 0; s < W_ / 4; ++s) {
    const int wb = s * 4 + kk;
    v2f a, bb;
    // A[m, wb..wb+1] = xp[b, k0+m, wb] = x[b, wb, k0+m]
    a.x = xb[wb * K_ + k0 + m];
    a.y = xb[(wb + 1) * K_ + k0 + m];
    // B[wb.., e0+m] = lin_w[e0+m, wOff+wb..]
    bb.x = wrow[wb];
    bb.y = wrow[wb + 1];
    c = __builtin_amdgcn_wmma_f32_16x16x4_f32(false, a, false, bb,
                                              (short)0, c, false, false);
  }

  // D layout: k = k0 + v + 8*(lane>>4), e = e0 + (lane&15)
  float* dst = (isR ? Rp : Lp) + ((b * K_ + k0) * E_) + e0 + m;
  const int rowAdd = (lane >> 4) * 8;
#pragma unroll
  for (int v = 0; v < 8; ++v) dst[(v + rowAdd) * E_] = c[v];
}

// ---------------------------------------------------------------------------
// Kernel B: e[b,i,j] = sum_e leaky(L[b,i,e]+R[b,j,e]+lin_b[e]) * a[e] + bias
//           att[b,i,:] = softmax_j
// Block = 128 threads = one j-row sweep for 8 consecutive i rows of batch b.
// Lb tiles staged in LDS (broadcast reads); R read once per (block, j, e):
// 8x reuse vs naive. Nonlinearity inside reduction => pure VALU stage.
// ---------------------------------------------------------------------------
__global__ __launch_bounds__(128) void gat_score_softmax_kernel(
    const float* __restrict__ Lp, const float* __restrict__ Rp,
    const float* __restrict__ lin_b, const float* __restrict__ av,
    const float* __restrict__ bias, float* __restrict__ att) {
  __shared__ float Lb[8 * E_];
  __shared__ float sa[E_];
  __shared__ float red[K_];

  const int tid = threadIdx.x;
  const int b   = blockIdx.x >> 4;
  const int i0  = (blockIdx.x & 15) << 3;

  for (int idx = tid; idx < 8 * E_; idx += 128) {
    const int ii = idx >> 8;
    const int e  = idx & (E_ - 1);
    Lb[idx] = Lp[((b * K_ + i0 + ii) * E_) + e] + lin_b[e];
  }
  for (int idx = tid; idx < E_; idx += 128) sa[idx] = av[idx];
  __syncthreads();

  const int j = tid;
  const float* rrow = Rp + (b * K_ + j) * E_;
  float acc[8] = {0.f, 0.f, 0.f, 0.f, 0.f, 0.f, 0.f, 0.f};
#pragma unroll 4
  for (int e = 0; e < E_; ++e) {
    const float r  = rrow[e];
    const float ae = sa[e];
#pragma unroll
    for (int ii = 0; ii < 8; ++ii) {
      const float z = Lb[ii * E_ + e] + r;
      const float w = (z >= 0.f) ? ae : NEG_SLOPE * ae;  // leaky(z)*a = z*w
      acc[ii] = fmaf(z, w, acc[ii]);
    }
  }

  float sc[8];
#pragma unroll
  for (int ii = 0; ii < 8; ++ii)
    sc[ii] = acc[ii] + bias[(i0 + ii) * K_ + j];

  float* attb = att + (b * K_ + i0) * K_;
  for (int ii = 0; ii < 8; ++ii) {
    red[j] = sc[ii];
    __syncthreads();
    for (int off = 64; off > 0; off >>= 1) {
      if (j < off) red[j] = fmaxf(red[j], red[j + off]);
      __syncthreads();
    }
    const float mx = red[0];
    __syncthreads();
    const float ex = __expf(sc[ii] - mx);
    red[j] = ex;
    __syncthreads();
    for (int off = 64; off > 0; off >>= 1) {
      if (j < off) red[j] += red[j + off];
      __syncthreads();
    }
    attb[ii * K_ + j] = ex / red[0];
    __syncthreads();
  }
}

// ---------------------------------------------------------------------------
// Kernel C: out[b,w,i] = sigmoid( sum_j att[b,i,j] * x[b,w,j] )
// One wave per 16x16 (i,w) tile, WMMA f32, inner dim j=K. Sigmoid fused on
// the accumulator, stored directly in the transposed (B,W,K) output layout.
// Tiles: b(32) * itile(8) * wtile(8) = 2048 waves.
// ---------------------------------------------------------------------------
__global__ __launch_bounds__(128) void gat_av_kernel(
    const float* __restrict__ att, const float* __restrict__ x,
    float* __restrict__ out) {
  const int wave = threadIdx.x >> 5;
  const int lane = threadIdx.x & 31;
  const int tile = blockIdx.x * 4 + wave;
  const int b  = tile >> 6;
  const int it = (tile >> 3) & 7;
  const int wt = tile & 7;
  const int i0 = it << 4;
  const int w0 = wt << 4;

  const int n  = lane & 15;
  const int kk = (lane >> 4) << 1;

  const float* attb = att + (b * K_ + i0 + n) * K_;       // A row i0+n
  const float* xb   = x + (b * W_ + w0 + n) * K_;         // B col w0+n: x[b,w,j]

  v8f c = {};
#pragma unroll 8
  for (int s = 0; s < K_ / 4; ++s) {
    const int jb = s * 4 + kk;
    v2f a, bb;
    a.x  = attb[jb];
    a.y  = attb[jb + 1];
    bb.x = xb[jb];
    bb.y = xb[jb + 1];
    c = __builtin_amdgcn_wmma_f32_16x16x4_f32(false, a, false, bb,
                                              (short)0, c, false, false);
  }

  // D: i = i0 + v + 8*(lane>>4), w = w0 + n ; store out[b, w, i] = sigmoid
  float* ob = out + ((b * W_ + w0 + n) * K_) + i0 + (lane >> 4) * 8;
#pragma unroll
  for (int v = 0; v < 8; ++v)
    ob[v] = 1.f / (1.f + __expf(-c[v]));
}

extern "C" void kernel_launch(void* const* d_in, const int* in_sizes, int n_in,
                              void* d_out, int out_size, void* d_ws, size_t ws_size,
                              hipStream_t stream) {
  (void)in_sizes; (void)n_in; (void)out_size; (void)ws_size;
  const float* x     = (const float*)d_in[0];   // (B, W, K)
  const float* lin_w = (const float*)d_in[1];   // (E, 2W)
  const float* lin_b = (const float*)d_in[2];   // (E,)
  const float* a     = (const float*)d_in[3];   // (E, 1)
  const float* bias  = (const float*)d_in[4];   // (K, K)
  float* out = (float*)d_out;                   // (B, W, K)

  float* Lp  = (float*)d_ws;                    // B*K*E floats (4 MB)
  float* Rp  = Lp + (size_t)B_ * K_ * E_;       // 4 MB
  float* att = Rp + (size_t)B_ * K_ * E_;       // B*K*K floats (2 MB)

  // A: 8192 tiles, 4 waves/block -> 2048 blocks
  gat_proj_kernel<<<2048, 128, 0, stream>>>(x, lin_w, Lp, Rp);
  // B: B * K/8 = 512 blocks
  gat_score_softmax_kernel<<<512, 128, 0, stream>>>(Lp, Rp, lin_b, a, bias, att);
  // C: 2048 tiles, 4 waves/block -> 512 blocks
  gat_av_kernel<<<512, 128, 0, stream>>>(att, x, out);
}